// SO3onS2_20005957665092
// MI455X (gfx1250) — compile-verified
//
#include <hip/hip_runtime.h>
#include <math.h>

typedef __attribute__((ext_vector_type(2))) float v2f;
typedef __attribute__((ext_vector_type(2), aligned(4))) float v2fu; // align-4 pair load
typedef __attribute__((ext_vector_type(8))) float v8f;

#define LMAXc 8
#define LLc   9           // LMAX+1
#define Wc    17          // 2*LMAX+1
#define FST   18          // padded f_padded stride, slot 17 == 0.0f
#define Dc    289         // W*W
#define Nc    215         // number of (l1,l2,l) triples
#define Bc    32          // batch
#define NLATc 64
#define NLONc 128
#define MTILES 19         // ceil(289/16)
#define KCHUNK 64
#define NCHUNKS 5         // 5*64 = 320 >= 289
#define KPAD  (NCHUNKS * KCHUNK)

// K-paired LDS layout for B: element (k, col) lives at
//   sB[(k>>1)*128 + col*2 + (k&1)]
// so the WMMA B fragment {k, k+1} (k always even here) is one aligned b64.
#define SB_IDX(k, col) ((((k) >> 1) * 128) + ((col) * 2) + ((k) & 1))

#if __has_builtin(__builtin_amdgcn_global_load_async_to_lds_b32) && \
    __has_builtin(__builtin_amdgcn_s_wait_asynccnt)
#define USE_ASYNC_LDS 1
typedef __attribute__((address_space(1))) int as1_int;
typedef __attribute__((address_space(3))) int as3_int;
#else
#define USE_ASYNC_LDS 0
#endif

// ---------------------------------------------------------------- zero output
__global__ void k_zero(float* __restrict__ out, int n) {
    int i = blockIdx.x * 256 + threadIdx.x;
    if (i < n) out[i] = 0.0f;
}

// ------------------------------------------------- DFT: fm[b,t,m], m = 0..8
__global__ void k_dft(const float* __restrict__ f,
                      float* __restrict__ fm_re, float* __restrict__ fm_im) {
    int idx = blockIdx.x * 256 + threadIdx.x;          // (b*NLAT + t)*LL + m
    if (idx >= Bc * NLATc * LLc) return;
    int m  = idx % LLc;
    int bt = idx / LLc;
    const float* row = f + (size_t)bt * NLONc;
    float w = -2.0f * 3.14159265358979323846f * (float)m / (float)NLONc;
    float sre = 0.0f, sim = 0.0f;
    for (int x = 0; x < NLONc; ++x) {
        float s, c;
        __sincosf(w * (float)x, &s, &c);
        float v = row[x];
        sre += v * c;
        sim += v * s;
    }
    fm_re[idx] = sre;
    fm_im[idx] = sim;
}

// ------------- Legendre contraction + mirror -> f_padded[b,l,mf], stride FST
// slot mf==17 is written as 0 so a sentinel gather index yields exact zero.
__global__ void k_fpad(const float* __restrict__ leg,
                       const float* __restrict__ fm_re, const float* __restrict__ fm_im,
                       float* __restrict__ fp_re, float* __restrict__ fp_im) {
    int idx = blockIdx.x * 256 + threadIdx.x;          // (b*LL + l)*FST + mf
    if (idx >= Bc * LLc * FST) return;
    int mf = idx % FST;
    int bl = idx / FST;
    int l  = bl % LLc;
    int b  = bl / LLc;
    float cr = 0.0f, ci = 0.0f;
    if (mf < Wc) {
        int m  = mf - LMAXc;
        int am = m < 0 ? -m : m;
        if (am <= l) {
            const float* lg = leg + ((size_t)l * LLc + am) * NLATc;
            for (int t = 0; t < NLATc; ++t) {
                float w = lg[t];
                size_t fi = ((size_t)b * NLATc + t) * LLc + am;
                cr += w * fm_re[fi];
                ci += w * fm_im[fi];
            }
            if (m < 0) {                                // sign * conj
                float s = (am & 1) ? -1.0f : 1.0f;
                cr = s * cr;
                ci = -s * ci;
            }
        }
    }
    fp_re[idx] = cr;
    fp_im[idx] = ci;
}

// ------------------------------------------------------------- main WMMA GEMM
// Per n: A = cg[n] (289x289 f32), B[e][col] = col<32 ? fhat_re[e][col]
//                                           : fhat_im[e][col-32]   (289x64)
// C = A*B gives P = cg@fhat_re, Q = cg@fhat_im.  With g = cg@conj(fhat):
//   g_re = P, g_im = -Q, and out = sum_d outer[d]*g[d]:
//   out_re = sum_d (o_re*P + o_im*Q),  out_im = sum_d (o_im*P - o_re*Q)
//
// No value masking needed in the GEMM:
//  - garbage A rows (row >= 289, address-clamped) only corrupt D rows >= 289,
//    which the epilogue never reads;
//  - garbage A columns (k >= 289, address-clamped) multiply B rows that were
//    staged as exact zeros via the gather sentinel.
__global__ void __launch_bounds__(256)
k_main(const float* __restrict__ cg,
       const int* __restrict__ l1a, const int* __restrict__ l2a,
       const int* __restrict__ la,
       const int* __restrict__ gidx, const unsigned char* __restrict__ gmask,
       const float* __restrict__ fp_re, const float* __restrict__ fp_im,
       float* __restrict__ out) {
    __shared__ float sB[KCHUNK * 64];                  // staged B chunk (16 KB)
    __shared__ int   s_gidx[KPAD];                     // sentinel = 17 -> zero slot
    __shared__ float sf1r[Bc * Wc], sf1i[Bc * Wc];
    __shared__ float sf2r[Bc * Wc], sf2i[Bc * Wc];

    const int n   = blockIdx.x;
    const int tid = threadIdx.x;
    const int l1  = l1a[n], l2 = l2a[n], l = la[n];

    // one-time: l1/l2 slices for the epilogue (stride Wc in LDS)
    for (int i = tid; i < Bc * Wc; i += 256) {
        int b = i / Wc, w = i % Wc;
        size_t base = (size_t)b * LLc;
        sf1r[i] = fp_re[(base + l1) * FST + w];
        sf1i[i] = fp_im[(base + l1) * FST + w];
        sf2r[i] = fp_re[(base + l2) * FST + w];
        sf2i[i] = fp_im[(base + l2) * FST + w];
    }
    // one-time: gather index table with zero-sentinel
    {
        const int*           gix = gidx  + (size_t)n * Dc;
        const unsigned char* gmk = gmask + (size_t)n * Dc;
        for (int e = tid; e < KPAD; e += 256) {
            int g = Wc;                                 // sentinel -> fp slot 17 == 0
            if (e < Dc && gmk[e]) g = gix[e];
            s_gidx[e] = g;
        }
    }

    const int lane  = tid & 31;
    const int wv    = tid >> 5;        // wave id 0..7 (wave32)
    const int lhalf = lane >> 4;       // 0 / 1
    const int lmod  = lane & 15;

    v8f acc[3][4];
#pragma unroll
    for (int ti = 0; ti < 3; ++ti)
#pragma unroll
        for (int nt = 0; nt < 4; ++nt)
            acc[ti][nt] = {};

    const float* cgn = cg + (size_t)n * Dc * Dc;

    for (int c = 0; c < NCHUNKS; ++c) {
        __syncthreads();
        // stage B chunk (branch-free gather, async direct-to-LDS when available)
        for (int i = tid; i < KCHUNK * 64; i += 256) {
            int ke  = i >> 6;
            int col = i & 63;
            int g   = s_gidx[c * KCHUNK + ke];
            int b   = col & 31;
            const float* src =
                ((col < 32) ? fp_re : fp_im) + ((size_t)b * LLc + l) * FST + g;
#if USE_ASYNC_LDS
            __builtin_amdgcn_global_load_async_to_lds_b32(
                (as1_int*)src, (as3_int*)&sB[SB_IDX(ke, col)], 0, 0);
#else
            sB[SB_IDX(ke, col)] = *src;
#endif
        }
#if USE_ASYNC_LDS
        __builtin_amdgcn_s_wait_asynccnt(0);
#endif
        __syncthreads();

        const bool tail = (c == NCHUNKS - 1);
        for (int ti = 0; ti < 3; ++ti) {
            int mt = wv + ti * 8;                      // wave-uniform
            if (mt >= MTILES) break;
            int r    = mt * 16 + lmod;
            int rowc = r < Dc ? r : (Dc - 1);          // address clamp only
            const float* arow = cgn + (size_t)rowc * Dc;
            // split K into two halves of 8 k-steps: abuf[8] keeps VGPR count
            // below 256 while still hiding the load clause behind 32 WMMAs
            for (int h = 0; h < 2; ++h) {
                v2f abuf[8];
                if (!tail) {
#pragma unroll
                    for (int kk = 0; kk < 8; ++kk) {
                        int kA = c * KCHUNK + (h * 8 + kk) * 4 + 2 * lhalf;
                        abuf[kk] = *(const v2fu*)(arow + kA);
                    }
                } else {
#pragma unroll
                    for (int kk = 0; kk < 8; ++kk) {
                        int kA = c * KCHUNK + (h * 8 + kk) * 4 + 2 * lhalf;
                        int k0 = kA     < Dc ? kA     : (Dc - 1);
                        int k1 = kA + 1 < Dc ? kA + 1 : (Dc - 1);
                        abuf[kk].x = arow[k0];         // junk when k>=289: B row = 0
                        abuf[kk].y = arow[k1];
                    }
                }
#pragma unroll
                for (int kk = 0; kk < 8; ++kk) {
                    int kB = (h * 8 + kk) * 4 + 2 * lhalf;   // even
#pragma unroll
                    for (int nt = 0; nt < 4; ++nt) {
                        // K-paired layout -> one aligned contiguous b64 per frag
                        v2f bf = *(const v2f*)&sB[SB_IDX(kB, nt * 16 + lmod)];
                        acc[ti][nt] = __builtin_amdgcn_wmma_f32_16x16x4_f32(
                            false, abuf[kk], false, bf, (short)0, acc[ti][nt],
                            false, false);
                    }
                }
            }
        }
    }
    __syncthreads();

    // epilogue: combine P/Q tiles with complex outer factor, atomically accumulate
    for (int ti = 0; ti < 3; ++ti) {
        int mt = wv + ti * 8;
        if (mt >= MTILES) break;
        for (int bh = 0; bh < 2; ++bh) {
            int b = bh * 16 + lmod;                    // batch column
            v8f P = acc[ti][bh];                       // cg @ fhat_re
            v8f Q = acc[ti][2 + bh];                   // cg @ fhat_im
            float are = 0.0f, aim = 0.0f;
#pragma unroll
            for (int v = 0; v < 8; ++v) {
                int d = mt * 16 + v + 8 * lhalf;       // C layout: M = v + 8*(lane/16)
                if (d < Dc) {
                    int i = d / Wc, j = d % Wc;
                    float x1r = sf1r[b * Wc + i], x1i = sf1i[b * Wc + i];
                    float x2r = sf2r[b * Wc + j], x2i = sf2i[b * Wc + j];
                    float ore = x1r * x2r - x1i * x2i;
                    float oim = x1r * x2i + x1i * x2r;
                    float p = P[v], q = Q[v];
                    are += ore * p + oim * q;
                    aim += oim * p - ore * q;
                }
            }
            atomicAdd(&out[((size_t)b * Nc + n) * 2 + 0], are);
            atomicAdd(&out[((size_t)b * Nc + n) * 2 + 1], aim);
        }
    }
}

// --------------------------------------------------------------------- launch
extern "C" void kernel_launch(void* const* d_in, const int* in_sizes, int n_in,
                              void* d_out, int out_size, void* d_ws, size_t ws_size,
                              hipStream_t stream) {
    const float*         f    = (const float*)d_in[0];
    const float*         leg  = (const float*)d_in[1];
    const float*         cg   = (const float*)d_in[2];
    const int*           l1a  = (const int*)d_in[3];
    const int*           l2a  = (const int*)d_in[4];
    const int*           la   = (const int*)d_in[5];
    const int*           gidx = (const int*)d_in[6];
    const unsigned char* gmk  = (const unsigned char*)d_in[7];
    float* out = (float*)d_out;

    // workspace layout (floats)
    float* ws    = (float*)d_ws;
    const int nfm = Bc * NLATc * LLc;                  // 18432
    const int nfp = Bc * LLc * FST;                    // 5184 (padded stride)
    float* fm_re = ws;
    float* fm_im = ws + nfm;
    float* fp_re = ws + 2 * nfm;
    float* fp_im = ws + 2 * nfm + nfp;

    const int nout = Bc * Nc * 2;                      // complex64 as float pairs
    k_zero<<<(nout + 255) / 256, 256, 0, stream>>>(out, nout);
    k_dft <<<(nfm + 255) / 256, 256, 0, stream>>>(f, fm_re, fm_im);
    k_fpad<<<(Bc * LLc * FST + 255) / 256, 256, 0, stream>>>(leg, fm_re, fm_im, fp_re, fp_im);
    k_main<<<Nc, 256, 0, stream>>>(cg, l1a, l2a, la, gidx, gmk, fp_re, fp_im, out);
    (void)in_sizes; (void)n_in; (void)out_size; (void)ws_size;
}